// IntraLevelMP_88751204204556
// MI455X (gfx1250) — compile-verified
//
#include <hip/hip_runtime.h>

// ============================================================================
// GNN message-passing layer for MI455X (gfx1250, wave32).
//
// Algorithmic hoist: message MLP depends only on x[src] -> compute per NODE
// (2.6 GFLOP) instead of per EDGE (42 GFLOP), then scatter-mean M[src]->agg[dst].
// M (20.5MB) + agg (20.5MB) stay resident in the 192MB L2, so the 640K-edge
// gather + 82M f32 atomics never hit HBM. fp32 WMMA (v_wmma_f32_16x16x4_f32)
// keeps the reference's fp32 numerics while using the CDNA5 matrix pipes.
//
// v3: weights (and the h1 slab) are stored in LDS with a per-8-column
// permutation [0,1,4,5 | 2,3,6,7] so one ds_load_b128 delivers a lane's
// B fragments for TWO k-steps. With two 16-row M-tiles per wave, each
// ds_load_b128 now feeds FOUR v_wmma (was: one ds op per two wmma).
// ============================================================================

typedef __attribute__((ext_vector_type(2))) float v2f;
typedef __attribute__((ext_vector_type(8))) float v8f;

#define NN  40000
#define NE  640000
#define D   128
#define LDA 132   // 128 + 4 pad: fragment rows land on distinct LDS banks

// Fragment layout (ISA 7.12.2): lane L<16 -> rows r0+L, k = k0,k0+1;
// lanes 16-31 -> same rows, k = k0+2,k0+3.
__device__ __forceinline__ v2f ld_frag_g(const float* __restrict__ P, int ld,
                                         int r0, int k0, int lane, int rmax) {
  int r = r0 + (lane & 15);
  r = r < rmax ? r : rmax - 1;
  int k = k0 + ((lane >> 4) << 1);
  v2f a;
  a.x = P[(size_t)r * ld + k];
  a.y = P[(size_t)r * ld + k + 1];
  return a;
}

#define WMMA_F32(A, B, C) \
  __builtin_amdgcn_wmma_f32_16x16x4_f32(false, (A), false, (B), (short)0, (C), false, false)

// Store a global float4 (row r, cols c..c+3, c%4==0) into a k-octet-swizzled
// LDS row: octet layout [k0,k1,k4,k5, k2,k3,k6,k7]. Then a lane (half h) reads
// ONE 16B ds_load_b128 at (row*LDA + (k&~7) + 4h) to get fragments for k-steps
// kb and kb+4.
__device__ __forceinline__ void store_swz(float* S, int ld, int r, int c,
                                          const float4 w) {
  float* p = S + r * ld + (c & ~7) + ((c & 4) ? 2 : 0);
  float2 lo, hi;
  lo.x = w.x; lo.y = w.y;   // k=c,c+1
  hi.x = w.z; hi.y = w.w;   // k=c+2,c+3
  *(float2*)(p)     = lo;
  *(float2*)(p + 4) = hi;
}

// Swizzled column index for a k within its octet (for epilogue stores).
__device__ __forceinline__ int swz_col(int k) {
  int p = (k >> 1) & 3;                  // pair id in octet
  int perm = ((p & 1) << 1) | (p >> 1);  // 0,2,1,3 -> slots 0,4,2,6
  return (k & ~7) | (perm << 1) | (k & 1);
}

// One b128 read = B fragments for k-steps kb and kb+4 (row = rbase+nlo).
__device__ __forceinline__ void ld_frag2_s(const float* S, int ld, int rbase,
                                           int kb, int nlo, int half,
                                           v2f* bl, v2f* bh) {
  const float4 q = *(const float4*)(S + (rbase + nlo) * ld + kb + 4 * half);
  bl->x = q.x; bl->y = q.y;
  bh->x = q.z; bh->y = q.w;
}

// ---------------------------------------------------------------------------
// Kernel 1: M = relu(x @ Wm1^T + bm1) @ Wm2^T + bm2   (per node, fused)
// 8 waves/block, 32 rows/wave (2 M-tiles), 256 rows/block.
// ---------------------------------------------------------------------------
__global__ __launch_bounds__(256)
void msg_mlp(const float* __restrict__ x,
             const float* __restrict__ Wm1, const float* __restrict__ bm1,
             const float* __restrict__ Wm2, const float* __restrict__ bm2,
             float* __restrict__ Mout)
{
  extern __shared__ float smem[];
  float* sW1 = smem;                  // D x LDA, k-swizzled
  float* sW2 = sW1 + D * LDA;         // D x LDA, k-swizzled
  float* sH  = sW2 + D * LDA;         // 256 x LDA, k-swizzled (wave-private)

  const int tid  = threadIdx.x;
  const int lane = tid & 31;
  const int wave = tid >> 5;
  const int r0   = blockIdx.x * 256 + wave * 32;
  const int half = lane >> 4;
  const int nlo  = lane & 15;
  const int snlo = swz_col(nlo);      // swizzled column offset for epilogues

  for (int i = tid * 4; i < D * D; i += 256 * 4) {
    int r = i >> 7, c = i & 127;
    store_swz(sW1, LDA, r, c, *(const float4*)(Wm1 + r * D + c));
    store_swz(sW2, LDA, r, c, *(const float4*)(Wm2 + r * D + c));
  }
  __syncthreads();

  const v8f vzero = {};
  v8f acc0[8], acc1[8];
#pragma unroll
  for (int j = 0; j < 8; ++j) { acc0[j] = vzero; acc1[j] = vzero; }

  // Layer 1: K=128, one octet (2 k-steps) per iteration.
  for (int kb = 0; kb < D; kb += 8) {
    v2f a00 = ld_frag_g(x, D, r0,      kb,     lane, NN);
    v2f a01 = ld_frag_g(x, D, r0,      kb + 4, lane, NN);
    v2f a10 = ld_frag_g(x, D, r0 + 16, kb,     lane, NN);
    v2f a11 = ld_frag_g(x, D, r0 + 16, kb + 4, lane, NN);
#pragma unroll
    for (int j = 0; j < 8; ++j) {
      v2f bl, bh;
      ld_frag2_s(sW1, LDA, j * 16, kb, nlo, half, &bl, &bh);  // 1 ds_load_b128
      acc0[j] = WMMA_F32(a00, bl, acc0[j]);                   // -> 4 wmma
      acc1[j] = WMMA_F32(a10, bl, acc1[j]);
      acc0[j] = WMMA_F32(a01, bh, acc0[j]);
      acc1[j] = WMMA_F32(a11, bh, acc1[j]);
    }
  }

  // bias + relu -> swizzled h1 slab (C-layout lane holds row v+8*half, col j*16+nlo).
#pragma unroll
  for (int j = 0; j < 8; ++j) {
    float bias = bm1[j * 16 + nlo];
#pragma unroll
    for (int v = 0; v < 8; ++v) {
      int m = v + 8 * half;
      sH[(wave * 32 + m) * LDA + j * 16 + snlo]      = fmaxf(acc0[j][v] + bias, 0.0f);
      sH[(wave * 32 + 16 + m) * LDA + j * 16 + snlo] = fmaxf(acc1[j][v] + bias, 0.0f);
    }
    acc0[j] = vzero;
    acc1[j] = vzero;
  }
  // slabs are wave-private: same-wave LDS RAW ordered by DScnt waits.

  // Layer 2: A fragments also come from swizzled LDS -> b128 per tile/octet.
  for (int kb = 0; kb < D; kb += 8) {
    v2f a00, a01, a10, a11;
    ld_frag2_s(sH, LDA, wave * 32,      kb, nlo, half, &a00, &a01);
    ld_frag2_s(sH, LDA, wave * 32 + 16, kb, nlo, half, &a10, &a11);
#pragma unroll
    for (int j = 0; j < 8; ++j) {
      v2f bl, bh;
      ld_frag2_s(sW2, LDA, j * 16, kb, nlo, half, &bl, &bh);
      acc0[j] = WMMA_F32(a00, bl, acc0[j]);
      acc1[j] = WMMA_F32(a10, bl, acc1[j]);
      acc0[j] = WMMA_F32(a01, bh, acc0[j]);
      acc1[j] = WMMA_F32(a11, bh, acc1[j]);
    }
  }

#pragma unroll
  for (int j = 0; j < 8; ++j) {
    float bias = bm2[j * 16 + nlo];
#pragma unroll
    for (int v = 0; v < 8; ++v) {
      int m = v + 8 * half;
      int row0 = r0 + m, row1 = r0 + 16 + m;
      if (row0 < NN) Mout[(size_t)row0 * D + j * 16 + nlo] = acc0[j][v] + bias;
      if (row1 < NN) Mout[(size_t)row1 * D + j * 16 + nlo] = acc1[j][v] + bias;
    }
  }
}

// ---------------------------------------------------------------------------
// Kernel 2: zero agg / cnt.
// ---------------------------------------------------------------------------
__global__ void zero_ws(float* __restrict__ agg, float* __restrict__ cnt) {
  int i = blockIdx.x * blockDim.x + threadIdx.x;
  int stride = gridDim.x * blockDim.x;
  for (int t = i; t < NN * D; t += stride) agg[t] = 0.0f;
  for (int t = i; t < NN; t += stride)     cnt[t] = 0.0f;
}

// ---------------------------------------------------------------------------
// Kernel 3: scatter-add. One wave per edge: float4 gather of M[src] (L2 hit),
// 4 atomicAdds/lane into agg[dst]; lane 0 bumps cnt[dst].
// ---------------------------------------------------------------------------
__global__ __launch_bounds__(256)
void scatter_mean_acc(const int* __restrict__ ei, const float* __restrict__ M,
                      float* __restrict__ agg, float* __restrict__ cnt)
{
  int t = blockIdx.x * 256 + threadIdx.x;
  int e = t >> 5, lane = t & 31;
  if (e >= NE) return;
  int src = ei[e];
  int dst = ei[NE + e];
  const float4 m4 = *(const float4*)(M + (size_t)src * D + lane * 4);
  float* a = agg + (size_t)dst * D + lane * 4;
  atomicAdd(a + 0, m4.x);
  atomicAdd(a + 1, m4.y);
  atomicAdd(a + 2, m4.z);
  atomicAdd(a + 3, m4.w);
  if (lane == 0) atomicAdd(cnt + dst, 1.0f);
}

// ---------------------------------------------------------------------------
// Kernel 4: u = relu([x | agg/cnt] @ Wu1^T + bu1) @ Wu2^T + bu2;
//           out = LayerNorm(u + x) * gamma + beta. Fully fused.
// The 256-row slab ALIASES Wu1's LDS region (Wu1 dead after layer 1;
// __syncthreads() separates last sW1 read from first slab write).
// ---------------------------------------------------------------------------
#define LDB 260                 // 256 + 4 pad for Wu1 (k-swizzled rows)
#define S0_FLOATS (256 * LDA)   // 33792 > D*LDB = 33280, covers both uses

__global__ __launch_bounds__(256)
void upd_mlp(const float* __restrict__ x,
             const float* __restrict__ agg, const float* __restrict__ cnt,
             const float* __restrict__ Wu1, const float* __restrict__ bu1,
             const float* __restrict__ Wu2, const float* __restrict__ bu2,
             const float* __restrict__ gamma, const float* __restrict__ beta,
             float* __restrict__ out)
{
  extern __shared__ float smem[];
  float* sW1 = smem;                  // D x LDB, k-swizzled (phase 1)
  float* sH  = smem;                  // 256 x LDA (phase 2, aliased)
  float* sW2 = smem + S0_FLOATS;      // D x LDA, k-swizzled

  const int tid  = threadIdx.x;
  const int lane = tid & 31;
  const int wave = tid >> 5;
  const int r0   = blockIdx.x * 256 + wave * 32;
  const int half = lane >> 4;
  const int nlo  = lane & 15;
  const int snlo = swz_col(nlo);

  for (int i = tid * 4; i < D * 2 * D; i += 256 * 4) {
    int r = i >> 8, c = i & 255;
    store_swz(sW1, LDB, r, c, *(const float4*)(Wu1 + r * 2 * D + c));
  }
  for (int i = tid * 4; i < D * D; i += 256 * 4) {
    int r = i >> 7, c = i & 127;
    store_swz(sW2, LDA, r, c, *(const float4*)(Wu2 + r * D + c));
  }
  __syncthreads();

  int rA0 = r0 + nlo;       rA0 = rA0 < NN ? rA0 : NN - 1;
  int rA1 = r0 + 16 + nlo;  rA1 = rA1 < NN ? rA1 : NN - 1;
  const float rv0 = 1.0f / fmaxf(cnt[rA0], 1.0f);   // clamped mean divisor
  const float rv1 = 1.0f / fmaxf(cnt[rA1], 1.0f);

  const v8f vzero = {};
  v8f acc0[8], acc1[8];
#pragma unroll
  for (int j = 0; j < 8; ++j) { acc0[j] = vzero; acc1[j] = vzero; }

  // Layer 1 over K=256: octets 0..15 from x, 16..31 from agg * (1/cnt).
  const int k2 = 2 * half;
  for (int kb = 0; kb < 2 * D; kb += 8) {
    v2f a00, a01, a10, a11;
    if (kb < D) {
      const float* p0 = x + (size_t)rA0 * D + kb + k2;
      const float* p1 = x + (size_t)rA1 * D + kb + k2;
      a00.x = p0[0]; a00.y = p0[1]; a01.x = p0[4]; a01.y = p0[5];
      a10.x = p1[0]; a10.y = p1[1]; a11.x = p1[4]; a11.y = p1[5];
    } else {
      const float* p0 = agg + (size_t)rA0 * D + (kb - D) + k2;
      const float* p1 = agg + (size_t)rA1 * D + (kb - D) + k2;
      a00.x = p0[0] * rv0; a00.y = p0[1] * rv0;
      a01.x = p0[4] * rv0; a01.y = p0[5] * rv0;
      a10.x = p1[0] * rv1; a10.y = p1[1] * rv1;
      a11.x = p1[4] * rv1; a11.y = p1[5] * rv1;
    }
#pragma unroll
    for (int j = 0; j < 8; ++j) {
      v2f bl, bh;
      ld_frag2_s(sW1, LDB, j * 16, kb, nlo, half, &bl, &bh);
      acc0[j] = WMMA_F32(a00, bl, acc0[j]);
      acc1[j] = WMMA_F32(a10, bl, acc1[j]);
      acc0[j] = WMMA_F32(a01, bh, acc0[j]);
      acc1[j] = WMMA_F32(a11, bh, acc1[j]);
    }
  }

  __syncthreads();   // all waves done reading sW1 -> safe to reuse as sH

#pragma unroll
  for (int j = 0; j < 8; ++j) {
    float bias = bu1[j * 16 + nlo];
#pragma unroll
    for (int v = 0; v < 8; ++v) {
      int m = v + 8 * half;
      sH[(wave * 32 + m) * LDA + j * 16 + snlo]      = fmaxf(acc0[j][v] + bias, 0.0f);
      sH[(wave * 32 + 16 + m) * LDA + j * 16 + snlo] = fmaxf(acc1[j][v] + bias, 0.0f);
    }
    acc0[j] = vzero;
    acc1[j] = vzero;
  }

  // Layer 2.
  for (int kb = 0; kb < D; kb += 8) {
    v2f a00, a01, a10, a11;
    ld_frag2_s(sH, LDA, wave * 32,      kb, nlo, half, &a00, &a01);
    ld_frag2_s(sH, LDA, wave * 32 + 16, kb, nlo, half, &a10, &a11);
#pragma unroll
    for (int j = 0; j < 8; ++j) {
      v2f bl, bh;
      ld_frag2_s(sW2, LDA, j * 16, kb, nlo, half, &bl, &bh);
      acc0[j] = WMMA_F32(a00, bl, acc0[j]);
      acc1[j] = WMMA_F32(a10, bl, acc1[j]);
      acc0[j] = WMMA_F32(a01, bh, acc0[j]);
      acc1[j] = WMMA_F32(a11, bh, acc1[j]);
    }
  }

  // u + bias back into the (now fully consumed) slab, LINEAR columns for LN.
#pragma unroll
  for (int j = 0; j < 8; ++j) {
    float bias = bu2[j * 16 + nlo];
#pragma unroll
    for (int v = 0; v < 8; ++v) {
      int m = v + 8 * half;
      sH[(wave * 32 + m) * LDA + j * 16 + nlo]      = acc0[j][v] + bias;
      sH[(wave * 32 + 16 + m) * LDA + j * 16 + nlo] = acc1[j][v] + bias;
    }
  }

  // Residual + LayerNorm: wave handles its 32 rows; 32 lanes x 4 cols each.
  const int c0 = lane * 4;
  for (int rr = 0; rr < 32; ++rr) {
    int row = r0 + rr;
    if (row >= NN) break;
    const float4 u4 = *(const float4*)(sH + (wave * 32 + rr) * LDA + c0);
    const float4 x4 = *(const float4*)(x + (size_t)row * D + c0);
    float h0 = u4.x + x4.x, h1 = u4.y + x4.y, h2 = u4.z + x4.z, h3 = u4.w + x4.w;
    float s  = h0 + h1 + h2 + h3;
    float ss = h0 * h0 + h1 * h1 + h2 * h2 + h3 * h3;
#pragma unroll
    for (int off = 16; off > 0; off >>= 1) {
      s  += __shfl_xor(s, off, 32);
      ss += __shfl_xor(ss, off, 32);
    }
    float mean = s * (1.0f / 128.0f);
    float var  = ss * (1.0f / 128.0f) - mean * mean;
    float inv  = rsqrtf(var + 1e-5f);
    const float4 g4 = *(const float4*)(gamma + c0);
    const float4 b4 = *(const float4*)(beta + c0);
    float4 o;
    o.x = (h0 - mean) * inv * g4.x + b4.x;
    o.y = (h1 - mean) * inv * g4.y + b4.y;
    o.z = (h2 - mean) * inv * g4.z + b4.z;
    o.w = (h3 - mean) * inv * g4.w + b4.w;
    *(float4*)(out + (size_t)row * D + c0) = o;
  }
}

// ---------------------------------------------------------------------------
extern "C" void kernel_launch(void* const* d_in, const int* in_sizes, int n_in,
                              void* d_out, int out_size, void* d_ws, size_t ws_size,
                              hipStream_t stream) {
  const float* x     = (const float*)d_in[0];
  const int*   ei    = (const int*)d_in[1];   // (2,E): [0..E)=src, [E..2E)=dst
  const float* Wm1   = (const float*)d_in[2];
  const float* bm1   = (const float*)d_in[3];
  const float* Wm2   = (const float*)d_in[4];
  const float* bm2   = (const float*)d_in[5];
  const float* Wu1   = (const float*)d_in[6];
  const float* bu1   = (const float*)d_in[7];
  const float* Wu2   = (const float*)d_in[8];
  const float* bu2   = (const float*)d_in[9];
  const float* gamma = (const float*)d_in[10];
  const float* beta  = (const float*)d_in[11];

  float* M   = (float*)d_ws;                  // N*D fp32 (20.5 MB)
  float* agg = M + (size_t)NN * D;            // N*D fp32 (20.5 MB)
  float* cnt = agg + (size_t)NN * D;          // N   fp32

  const size_t smem_msg = (size_t)(2 * D * LDA + 256 * LDA) * sizeof(float); // 270,336 B
  const size_t smem_upd = (size_t)(S0_FLOATS + D * LDA) * sizeof(float);     // 202,752 B

  (void)hipFuncSetAttribute(reinterpret_cast<const void*>(msg_mlp),
                            hipFuncAttributeMaxDynamicSharedMemorySize, (int)smem_msg);
  (void)hipFuncSetAttribute(reinterpret_cast<const void*>(upd_mlp),
                            hipFuncAttributeMaxDynamicSharedMemorySize, (int)smem_upd);

  const int nodeBlocks = (NN + 255) / 256;    // 157

  msg_mlp<<<nodeBlocks, 256, smem_msg, stream>>>(x, Wm1, bm1, Wm2, bm2, M);
  zero_ws<<<1024, 256, 0, stream>>>(agg, cnt);
  scatter_mean_acc<<<(NE * 32) / 256, 256, 0, stream>>>(ei, M, agg, cnt);
  upd_mlp<<<nodeBlocks, 256, smem_upd, stream>>>(x, agg, cnt, Wu1, bu1, Wu2, bu2,
                                                 gamma, beta, (float*)d_out);
}